// st_attention_block_19043884990497
// MI455X (gfx1250) — compile-verified
//
#include <hip/hip_runtime.h>
#include <hip/hip_bf16.h>

// Dims: N=256, C=64, T=64, V=25, O=64, R=4
#define DN 256
#define DC 64
#define DT 64
#define DV 25
#define DO 64

typedef float v2f __attribute__((ext_vector_type(2)));
typedef float v8f __attribute__((ext_vector_type(8)));

#if __has_builtin(__builtin_amdgcn_global_load_async_to_lds_b128)
#define HAVE_ASYNC_LDS 1
typedef int v4i_vec __attribute__((vector_size(4 * sizeof(int))));
typedef __attribute__((address_space(1))) v4i_vec gv4i;
typedef __attribute__((address_space(3))) v4i_vec lv4i;
#endif

__device__ __forceinline__ void wait_async_then_barrier() {
#if __has_builtin(__builtin_amdgcn_s_wait_asynccnt)
    __builtin_amdgcn_s_wait_asynccnt(0);
#else
    asm volatile("s_wait_asynccnt 0x0" ::: "memory");
#endif
    __syncthreads();
}

// -------------------------------------------------------------------------
// Kernel A: per-n -> temporal mean, k/q, tanh attention, score (to d_ws)
// score[n][o][i][j] = alpha * (wf @ tanh([w1@(k*q); w2@(k-q)]) + bf) + A[i][j]
// -------------------------------------------------------------------------
__global__ __launch_bounds__(256) void st_score_kernel(
    const float* __restrict__ x,  const float* __restrict__ A,
    const float* __restrict__ wk, const float* __restrict__ bk,
    const float* __restrict__ wq, const float* __restrict__ bq,
    const float* __restrict__ w1, const float* __restrict__ b1,
    const float* __restrict__ w2, const float* __restrict__ b2,
    const float* __restrict__ wf, const float* __restrict__ bf,
    const float* __restrict__ alpha, float* __restrict__ score)
{
    const int n   = blockIdx.x;
    const int tid = threadIdx.x;

    __shared__ float xm_s[DC * 26];      // stride 26 vs 64 banks
    __shared__ float k_s[4 * DV];
    __shared__ float q_s[4 * DV];
    __shared__ float att1[4 * DV * DV];
    __shared__ float att2[4 * DV * DV];

    // ---- 1. temporal mean: xm[c][v] = mean_t x[n,c,t,v] ----
    const float* xn = x + (size_t)n * (DC * DT * DV);
    float acc[7];
    int   off[7];
    int   np = 0;
    for (int p = tid; p < DC * DV; p += 256) {
        off[np] = (p / DV) * (DT * DV) + (p % DV);
        acc[np] = 0.f;
        ++np;
    }
    for (int t = 0; t < DT; ++t) {
        const int tb = t * DV;
        #pragma unroll 7
        for (int u = 0; u < np; ++u) acc[u] += xn[off[u] + tb];
    }
    {
        int u = 0;
        for (int p = tid; p < DC * DV; p += 256)
            xm_s[(p / DV) * 26 + (p % DV)] = acc[u++] * (1.f / (float)DT);
    }
    __syncthreads();

    // ---- 2. k[r][v], q[r][v] (R=4, V=25 -> 100 values) ----
    if (tid < 4 * DV) {
        const int r = tid / DV, v = tid % DV;
        float sk = bk[r], sq = bq[r];
        for (int c = 0; c < DC; ++c) {
            const float xv = xm_s[c * 26 + v];
            sk += wk[r * DC + c] * xv;
            sq += wq[r * DC + c] * xv;
        }
        k_s[r * DV + v] = sk;
        q_s[r * DV + v] = sq;
    }
    __syncthreads();

    // ---- 3. att = tanh([w1@(k_i*q_j); w2@(k_i-q_j)]) ----
    for (int e = tid; e < 4 * DV * DV; e += 256) {
        const int s = e / (DV * DV), ij = e % (DV * DV);
        const int i = ij / DV, j = ij % DV;
        float t1 = b1[s], t2 = b2[s];
        #pragma unroll
        for (int r = 0; r < 4; ++r) {
            const float kv = k_s[r * DV + i], qv = q_s[r * DV + j];
            t1 += w1[s * 4 + r] * (kv * qv);
            t2 += w2[s * 4 + r] * (kv - qv);
        }
        att1[s * (DV * DV) + ij] = tanhf(t1);
        att2[s * (DV * DV) + ij] = tanhf(t2);
    }
    __syncthreads();

    // ---- 4. score = alpha*(wf@att + bf) + A ----
    const float al = alpha[0];
    float* sc = score + (size_t)n * (DO * DV * DV);
    for (int f = tid; f < DO * DV * DV; f += 256) {
        const int o = f / (DV * DV), ij = f % (DV * DV);
        float s = bf[o];
        #pragma unroll
        for (int c = 0; c < 4; ++c) s += wf[o * 8 + c]     * att1[c * (DV * DV) + ij];
        #pragma unroll
        for (int c = 0; c < 4; ++c) s += wf[o * 8 + 4 + c] * att2[c * (DV * DV) + ij];
        sc[f] = s * al + A[ij];
    }
}

// -------------------------------------------------------------------------
// Kernel B: per (t-chunk of 16, n).
// Phase 1 (WMMA): v[o, t16, j] = wv(64x64) @ x(64 x 400) + bv  -> LDS
// Phase 2 (WMMA): out[o, t, i] = sum_j score[o,i,j] * v[o,t,j]
// fp32 WMMA 16x16x4; A: m=lane%16, {k0+2*half, +1}; B/D: col=lane%16.
// x staged with GLOBAL_LOAD_ASYNC_TO_LDS_B128 (ASYNCcnt) when available.
// -------------------------------------------------------------------------
#define XS 404            // LDS row stride for 400-column tiles (16B aligned)
#define WVS 65            // LDS row stride for wv (kills bank conflicts)

__global__ __launch_bounds__(256) void st_out_kernel(
    const float* __restrict__ x,  const float* __restrict__ wv,
    const float* __restrict__ bv, const float* __restrict__ score,
    float* __restrict__ out)
{
    extern __shared__ float sm[];
    float* wv_s = sm;                        // 64*65   = 4160
    float* bv_s = wv_s + DO * WVS;           // 64
    float* x_s  = bv_s + DO;                 // 64*404  = 25856  x[c][t*25+j]
    float* v_s  = x_s + DC * XS;             // 64*404  = 25856  v[o][t*25+j]

    const int tc   = blockIdx.x;             // t-chunk (0..3), 16 frames each
    const int n    = blockIdx.y;
    const int tid  = threadIdx.x;
    const int lane = tid & 31;
    const int wave = tid >> 5;               // wave32: 8 waves / 256 threads
    const int half = lane >> 4;
    const int lx   = lane & 15;

    // ---- stage wv, bv (through VGPRs; small + stride-65 layout) ----
    for (int f = tid; f < DO * DC; f += 256)
        wv_s[(f >> 6) * WVS + (f & 63)] = wv[f];
    if (tid < DO) bv_s[tid] = bv[tid];

    // ---- stage x chunk: 64 rows x 400 contiguous floats ----
    const float* xb = x + (size_t)n * (DC * DT * DV) + (size_t)tc * 16 * DV;
#if defined(HAVE_ASYNC_LDS)
    for (int f4 = tid; f4 < DC * 100; f4 += 256) {       // 6400 float4s
        const int c = f4 / 100, r4 = f4 % 100;
        v4i_vec* gp = (v4i_vec*)(xb + c * (DT * DV) + r4 * 4);
        v4i_vec* lp = (v4i_vec*)(x_s + c * XS + r4 * 4);
        __builtin_amdgcn_global_load_async_to_lds_b128(
            (gv4i*)gp, (lv4i*)lp, 0, 0);
    }
    wait_async_then_barrier();
#else
    for (int f4 = tid; f4 < DC * 100; f4 += 256) {
        const int c = f4 / 100, r4 = f4 % 100;
        *(float4*)(x_s + c * XS + r4 * 4) =
            *(const float4*)(xb + c * (DT * DV) + r4 * 4);
    }
    __syncthreads();
#endif

    // ---- Phase 1: v = wv @ x + bv.  Tiles: 4 (o) x 25 (t*j cols), K=64 ----
    for (int tt = wave; tt < 100; tt += 8) {
        const int ob = tt / 25, cb = tt % 25;
        const int m  = ob * 16 + lx;         // A row
        const int nn = cb * 16 + lx;         // B/D col
        v8f acc = {};
        #pragma unroll
        for (int k0 = 0; k0 < DC; k0 += 4) {
            const int ka = k0 + 2 * half;
            v2f a, b;
            a.x = wv_s[m * WVS + ka];
            a.y = wv_s[m * WVS + ka + 1];
            b.x = x_s[ka * XS + nn];
            b.y = x_s[(ka + 1) * XS + nn];
            acc = __builtin_amdgcn_wmma_f32_16x16x4_f32(
                false, a, false, b, (short)0, acc, false, false);
        }
        #pragma unroll
        for (int vg = 0; vg < 8; ++vg) {
            const int mo = ob * 16 + vg + 8 * half;   // D row
            v_s[mo * XS + nn] = acc[vg] + bv_s[mo];
        }
    }
    __syncthreads();

    // ---- Phase 2: out[t,i] = v[t,:] @ score[o,i,:]^T.  128 tiles (64 o x 2 i) ----
    const float* sc = score + (size_t)n * (DO * DV * DV);
    for (int tt = wave; tt < 128; tt += 8) {
        const int o  = tt >> 1, ib = tt & 1;
        const int i  = ib * 16 + lx;                  // B/D col
        const int ic = (i < DV - 1) ? i : (DV - 1);   // clamp OOB cols (masked at store)
        v8f acc = {};
        #pragma unroll
        for (int j0 = 0; j0 < 28; j0 += 4) {          // K = j, 25 padded to 28
            const int ja = j0 + 2 * half;
            v2f a, b;
            a.x = (ja     < DV) ? v_s[o * XS + lx * DV + ja]     : 0.f;
            a.y = (ja + 1 < DV) ? v_s[o * XS + lx * DV + ja + 1] : 0.f;
            const int j0c = (ja     < DV - 1) ? ja     : (DV - 1);
            const int j1c = (ja + 1 < DV - 1) ? ja + 1 : (DV - 1);
            b.x = sc[(o * DV + ic) * DV + j0c];       // A lane is 0 where clamped
            b.y = sc[(o * DV + ic) * DV + j1c];
            acc = __builtin_amdgcn_wmma_f32_16x16x4_f32(
                false, a, false, b, (short)0, acc, false, false);
        }
        if (i < DV) {
            float* op = out + (((size_t)n * DO + o) * DT + tc * 16) * DV + i;
            #pragma unroll
            for (int vg = 0; vg < 8; ++vg) {
                const int t = vg + 8 * half;          // D row = frame in chunk
                op[t * DV] = acc[vg];
            }
        }
    }
}

// -------------------------------------------------------------------------
extern "C" void kernel_launch(void* const* d_in, const int* in_sizes, int n_in,
                              void* d_out, int out_size, void* d_ws, size_t ws_size,
                              hipStream_t stream) {
    const float* x     = (const float*)d_in[0];
    const float* A     = (const float*)d_in[1];
    const float* wk    = (const float*)d_in[2];
    const float* bk    = (const float*)d_in[3];
    const float* wq    = (const float*)d_in[4];
    const float* bq    = (const float*)d_in[5];
    const float* wv    = (const float*)d_in[6];
    const float* bv    = (const float*)d_in[7];
    const float* w1    = (const float*)d_in[8];
    const float* b1    = (const float*)d_in[9];
    const float* w2    = (const float*)d_in[10];
    const float* b2    = (const float*)d_in[11];
    const float* wf    = (const float*)d_in[12];
    const float* bf    = (const float*)d_in[13];
    const float* alpha = (const float*)d_in[14];
    float*       out   = (float*)d_out;
    float*       score = (float*)d_ws;   // N*O*V*V f32 = 40.96 MB scratch

    st_score_kernel<<<DN, 256, 0, stream>>>(
        x, A, wk, bk, wq, bq, w1, b1, w2, b2, wf, bf, alpha, score);

    const size_t smem = (size_t)(DO * WVS + DO + 2 * DC * XS) * sizeof(float);
    st_out_kernel<<<dim3(DT / 16, DN), 256, smem, stream>>>(
        x, wv, bv, score, out);
}